// QuantumFeatureExtractor_65481071402938
// MI455X (gfx1250) — compile-verified
//
#include <hip/hip_runtime.h>

typedef unsigned int v4u __attribute__((ext_vector_type(4)));
typedef int          v8i __attribute__((ext_vector_type(8)));
typedef int          v4i __attribute__((ext_vector_type(4)));
typedef float        vf4 __attribute__((ext_vector_type(4)));

#define N_QUBITS        8
#define EMBED           768
#define ROWS_PER_BLOCK  256           // 256 rows * 32B = 8 KiB LDS tile per block
#define THREADS         256           // 8 waves (wave32)
#define ROWS_PER_WAVE   (ROWS_PER_BLOCK / 8)      // 32 rows -> 1 KiB LDS slice per wave
#define VECS_PER_WAVE   (ROWS_PER_WAVE * N_QUBITS / 4)  // 64 float4s per wave
#define ITERS           (VECS_PER_WAVE / 32)      // 2 float4s per lane

__global__ __launch_bounds__(THREADS)
void qfe_tdm_kernel(const float* __restrict__ x,
                    const float* __restrict__ theta,
                    float* __restrict__ out)
{
    __shared__ float tile[ROWS_PER_BLOCK * N_QUBITS];   // dense 256x8 f32 tile

    const unsigned int lane   = threadIdx.x & 31u;
    const unsigned int waveId = __builtin_amdgcn_readfirstlane(threadIdx.x >> 5);  // scalar

    const unsigned long long rowStart =
        (unsigned long long)blockIdx.x * (unsigned long long)ROWS_PER_BLOCK;
    const unsigned long long waveRow = rowStart + (unsigned long long)waveId * ROWS_PER_WAVE;

    // Workgroup-relative LDS byte offset of this wave's slice (low 32 bits of
    // the generic LDS-aperture address); ptrtoint also captures the allocation.
    const unsigned int ldsOff =
        (unsigned int)(unsigned long long)(uintptr_t)tile +
        waveId * (ROWS_PER_WAVE * N_QUBITS * 4);

    // ---- Each wave issues its own TDM: x[waveRow .. waveRow+32, 0:8] -> LDS slice ----
    {
        unsigned long long gaddr =
            (unsigned long long)(const void*)x + waveRow * (unsigned long long)(EMBED * 4);

        // D# group 0: [1:0]=count=1 | lds_addr | global_addr[56:0] | type=2
        v4u g0;
        g0[0] = 1u;                                   // count=1, user mode
        g0[1] = ldsOff;                               // lds_addr = this wave's slice
        g0[2] = (unsigned int)gaddr;                  // global_addr[31:0]
        g0[3] = ((unsigned int)(gaddr >> 32) & 0x01FFFFFFu) | (2u << 30); // addr[56:32] | type=2

        // D# group 1 (256b): wg_mask=0, data_size=4B, dims/strides in elements
        v8i g1;
        g1[0] = (int)(2u << 16);                            // data_size = 2 -> 4 bytes
        g1[1] = (int)((EMBED & 0xFFFFu) << 16);             // tensor_dim0 lo16 = 768
        g1[2] = (int)((ROWS_PER_WAVE & 0xFFFFu) << 16);     // dim0 hi16=0 | tensor_dim1 lo16 = 32
        g1[3] = (int)((unsigned)N_QUBITS << 16);            // dim1 hi16=0 | tile_dim0 = 8
        g1[4] = (int)ROWS_PER_WAVE;                         // tile_dim1 = 32 | tile_dim2 = 0
        g1[5] = (int)EMBED;                                 // tensor_dim0_stride lo32 = 768
        g1[6] = 0;                                          // stride hi16 = 0 | dim1_stride lo16 = 0
        g1[7] = 0;                                          // dim1_stride hi32 = 0

        v4i z4 = (v4i){0, 0, 0, 0};                         // groups 2/3: zero (2-D tensor)
        v8i z8 = (v8i){0, 0, 0, 0, 0, 0, 0, 0};
        __builtin_amdgcn_tensor_load_to_lds(g0, g1, z4, z4, z8, 0);
    }

    // ---- Overlap with DMA: cos(theta[0..7]) (uniform -> SALU float pipe) ----
    const vf4* th4 = (const vf4*)theta;
    vf4 th0 = th4[0];
    vf4 th1 = th4[1];
    vf4 ct0, ct1;
#pragma unroll
    for (int k = 0; k < 4; ++k) {
        ct0[k] = cosf(th0[k]);
        ct1[k] = cosf(th1[k]);
    }

    // Wait for THIS wave's DMA only (TENSORcnt is per-wave); no barrier needed
    // since each wave reads only the slice it loaded itself.
    __builtin_amdgcn_s_wait_tensorcnt(0);

    // Compiler fence: LDS was written by the TDM (opaque to the optimizer).
    asm volatile("" : : "r"(tile) : "memory");

    // ---- Stream LDS slice -> cos -> coalesced NT b128 stores ----
    const vf4* t4   = (const vf4*)tile;
    vf4*       out4 = ((vf4*)out) + rowStart * (N_QUBITS / 4);

    const vf4 c = (lane & 1u) ? ct1 : ct0;                  // half-row parity (stride-2 invariant)
#pragma unroll
    for (int i = 0; i < ITERS; ++i) {
        const int v = (int)(waveId * VECS_PER_WAVE + i * 32 + lane);
        vf4 a = t4[v];                                      // ds_load_b128, conflict-free
        vf4 r;
#pragma unroll
        for (int k = 0; k < 4; ++k) {
            r[k] = cosf(a[k]) * c[k];
        }
        __builtin_nontemporal_store(r, &out4[v]);           // global_store_b128 NT
    }
}

extern "C" void kernel_launch(void* const* d_in, const int* in_sizes, int n_in,
                              void* d_out, int out_size, void* d_ws, size_t ws_size,
                              hipStream_t stream) {
    const float* x     = (const float*)d_in[0];
    const float* theta = (const float*)d_in[1];
    float*       out   = (float*)d_out;

    const int totalRows = in_sizes[0] / EMBED;              // 16*4096 = 65536
    const int blocks    = totalRows / ROWS_PER_BLOCK;       // 256

    qfe_tdm_kernel<<<blocks, THREADS, 0, stream>>>(x, theta, out);
}